// Transition_48601849922239
// MI455X (gfx1250) — compile-verified
//
#include <hip/hip_runtime.h>
#include <math.h>

// ---------------------------------------------------------------------------
// Morphological gradient of sigmoid(x) with 5x5 elliptical SE (3x5 block + center column), threshold 0.5.
// Memory-bound: 268 MB traffic -> ~11.5 us roofline on MI455X (23.3 TB/s).
// Strategy: async global->LDS tile staging (CDNA5 ASYNCcnt path), separable
// max/min (H5 then V3, plus V5 center column), sigmoid applied only to the
// final max/min (monotonicity), clamp-to-edge == geodesic for this SE.
// Accurate expf (not v_exp approx) to keep the 0.5-threshold faithful; we
// have ~20x compute headroom under the memory roofline so it is free.
// ---------------------------------------------------------------------------

#define IMG 2048
#define TW 64
#define TH 32
#define RAW_H (TH + 4)         // 36 rows  (vertical halo 2)
#define RAW_W (TW + 4)         // 68 cols  (horizontal halo 2)
#define RAW_STRIDE (RAW_W + 1) // 69, pad to stagger LDS banks
#define H_H (TH + 2)           // 34 rows of horizontal-pass results

#if __has_builtin(__builtin_amdgcn_global_load_async_to_lds_b32)
#define HAVE_ASYNC_LDS 1
#else
#define HAVE_ASYNC_LDS 0
#endif

// Pointer types matching the builtin's parameters (AS1 int* / AS3 int*).
typedef __attribute__((address_space(1))) int global_int_t;
typedef __attribute__((address_space(3))) int lds_int_t;

__device__ __forceinline__ float sigmoid_f(float v) {
    return 1.0f / (1.0f + expf(-v));
}

__device__ __forceinline__ int clampi(int v, int lo, int hi) {
    return v < lo ? lo : (v > hi ? hi : v);
}

__global__ __launch_bounds__(256) void morph_grad_kernel(const float* __restrict__ x,
                                                         float* __restrict__ out) {
    __shared__ float raw[RAW_H * RAW_STRIDE];
    __shared__ float hmx[H_H * TW];
    __shared__ float hmn[H_H * TW];

    const int tid = threadIdx.x;
    const int c0 = blockIdx.x * TW;
    const int r0 = blockIdx.y * TH;
    const float* img = x + (size_t)blockIdx.z * ((size_t)IMG * IMG);
    float* oimg = out + (size_t)blockIdx.z * ((size_t)IMG * IMG);

    // ---- Stage raw tile + halo(2) into LDS with clamp-to-edge addressing ----
    // Clamp-to-edge is equivalent to kornia's geodesic padding for this SE:
    // every clamped tap lands on an in-bounds tap of the SE, and max/min over
    // a multiset equals max/min over its underlying set.
    for (int idx = tid; idx < RAW_H * RAW_W; idx += 256) {
        const int rr = idx / RAW_W;
        const int cc = idx - rr * RAW_W;
        const int gr = clampi(r0 + rr - 2, 0, IMG - 1);
        const int gc = clampi(c0 + cc - 2, 0, IMG - 1);
#if HAVE_ASYNC_LDS
        __builtin_amdgcn_global_load_async_to_lds_b32(
            (global_int_t*)(img + (size_t)gr * IMG + gc),
            (lds_int_t*)&raw[rr * RAW_STRIDE + cc],
            0, 0);
#else
        raw[rr * RAW_STRIDE + cc] = img[(size_t)gr * IMG + gc];
#endif
    }
#if HAVE_ASYNC_LDS
#if __has_builtin(__builtin_amdgcn_s_wait_asynccnt)
    __builtin_amdgcn_s_wait_asynccnt(0);
#else
    asm volatile("s_wait_asynccnt 0" ::: "memory");
#endif
#endif
    __syncthreads();

    // ---- Horizontal pass: 5-wide max/min over rows [r0-1 .. r0+TH] ----
    // H row hr corresponds to image row (r0 + hr - 1), i.e. raw row (hr + 1).
    for (int idx = tid; idx < H_H * TW; idx += 256) {
        const int hr = idx >> 6;       // / TW
        const int hc = idx & (TW - 1); // % TW
        const float* rp = &raw[(hr + 1) * RAW_STRIDE + hc];
        const float a0 = rp[0], a1 = rp[1], a2 = rp[2], a3 = rp[3], a4 = rp[4];
        hmx[idx] = fmaxf(fmaxf(fmaxf(a0, a1), fmaxf(a2, a3)), a4);
        hmn[idx] = fminf(fminf(fminf(a0, a1), fminf(a2, a3)), a4);
    }
    __syncthreads();

    // ---- Vertical pass + center-column V5 + sigmoid + threshold ----
    for (int idx = tid; idx < TH * TW; idx += 256) {
        const int orr = idx >> 6;      // output row within tile
        const int oc  = idx & (TW - 1);

        // V3 over H-pass results (image rows orr-1..orr+1 -> H rows orr..orr+2)
        float M = fmaxf(fmaxf(hmx[orr * TW + oc], hmx[(orr + 1) * TW + oc]),
                        hmx[(orr + 2) * TW + oc]);
        float m = fminf(fminf(hmn[orr * TW + oc], hmn[(orr + 1) * TW + oc]),
                        hmn[(orr + 2) * TW + oc]);

        // V5 on raw center column (image rows orr-2..orr+2 -> raw rows orr..orr+4)
        const float* cp = &raw[orr * RAW_STRIDE + (oc + 2)];
        const float v0 = cp[0 * RAW_STRIDE], v1 = cp[1 * RAW_STRIDE],
                    v2 = cp[2 * RAW_STRIDE], v3 = cp[3 * RAW_STRIDE],
                    v4 = cp[4 * RAW_STRIDE];
        M = fmaxf(M, fmaxf(fmaxf(v0, v1), fmaxf(fmaxf(v2, v3), v4)));
        m = fminf(m, fminf(fminf(v0, v1), fminf(fminf(v2, v3), v4)));

        // sigmoid is monotone: dil = sigmoid(M), ero = sigmoid(m)
        const float g = sigmoid_f(M) - sigmoid_f(m);
        oimg[(size_t)(r0 + orr) * IMG + (c0 + oc)] = (g > 0.5f) ? 1.0f : 0.0f;
    }
}

extern "C" void kernel_launch(void* const* d_in, const int* in_sizes, int n_in,
                              void* d_out, int out_size, void* d_ws, size_t ws_size,
                              hipStream_t stream) {
    (void)n_in; (void)out_size; (void)d_ws; (void)ws_size;
    const float* x = (const float*)d_in[0];
    float* out = (float*)d_out;
    const int nimg = in_sizes[0] / (IMG * IMG); // B*C images (8 for reference)
    dim3 grid(IMG / TW, IMG / TH, nimg);
    morph_grad_kernel<<<grid, 256, 0, stream>>>(x, out);
}